// WFVAE_85796266705018
// MI455X (gfx1250) — compile-verified
//
#include <hip/hip_runtime.h>

// ---------------------------------------------------------------------------
// 3-level 2D Haar approximation == 0.125 * (8x8 block sum).
// Input  x  : [2,3,16,1024,1024] f32  (96 planes of 1024x1024)
// Output out: [2,3,16, 128, 128] f32  (96 planes of 128x128)
//
// Memory-bound (384 MB read -> ~16.7 us @ 23.3 TB/s). Column pooling is done
// with V_WMMA_F32_16X16X4_F32 (D = A*B + C):
//   A (16x4)  = 16 input rows x 4 input cols  (lane = row, lane-half = K pair)
//   B (4x16)  = ones(4,1) * e_g^T, g = (4k)/8  -> every row of B identical,
//               so each lane's B value is just (lane%16 == k/2), layout-proof.
// After 32 K-chunks, D[m,n] = sum over 8 input cols (group n) of row m.
// Row pooling (8:1) is free from the C/D layout: VGPR v holds M=v (lanes 0-15)
// and M=v+8 (lanes 16-31), so sum(acc[0..7]) per lane gives the two output
// rows directly. One wave: 16x128 input tile -> 2x16 output tile.
// ---------------------------------------------------------------------------

typedef __attribute__((ext_vector_type(2))) float v2f;
typedef __attribute__((ext_vector_type(8))) float v8f;

#define PLANES     96
#define IN_W       1024
#define IN_H       1024
#define OUT_W      128
#define TILES_PP   512   // (1024/16 row tiles) * (1024/128 col tiles) = 64*8

__global__ __launch_bounds__(256)
void haar3_wmma_kernel(const float* __restrict__ x, float* __restrict__ out) {
    const int lane = threadIdx.x;        // 0..31 (wave32)
    const int m    = lane & 15;          // A-matrix row / D column index
    const int h    = lane >> 4;          // lane half: selects K pair / out row

    // global wave id -> (plane, row-tile, col-tile); exactly 96*512 waves
    const int wid = blockIdx.x * blockDim.y + threadIdx.y;
    const int p   = wid >> 9;            // plane 0..95
    const int t   = wid & (TILES_PP - 1);
    const int tr  = t >> 3;              // row tile 0..63  (16 input rows)
    const int tc  = t & 7;               // col tile 0..7   (128 input cols)

    // This lane's input row pointer in A layout:
    //   row = tr*16 + (lane%16), starting col = tc*128 + 2*(lane/16)
    const float* rowp = x + (size_t)p * ((size_t)IN_H * IN_W)
                          + (size_t)(tr * 16 + m) * IN_W
                          + (size_t)(tc * 128 + 2 * h);

    v8f acc = {};
#pragma unroll
    for (int k = 0; k < 32; ++k) {
        // A chunk: 2 consecutive f32 of this lane's row (8B aligned)
        v2f a = *(const v2f*)(rowp + 4 * k);
        // B chunk: all 4 rows == e_g, g = k/2 ; value depends only on column
        const float bv = (m == (k >> 1)) ? 1.0f : 0.0f;
        v2f b; b.x = bv; b.y = bv;
        // D(16x16) += A(16x4) * B(4x16)
        acc = __builtin_amdgcn_wmma_f32_16x16x4_f32(
            /*neg_a=*/false, a, /*neg_b=*/false, b,
            /*c_mod=*/(short)0, acc, /*reuse_a=*/false, /*reuse_b=*/false);
    }

    // Row pooling 8:1 straight out of the C/D layout:
    //   lanes 0-15 : sum of M=0..7  -> output row 2*tr
    //   lanes16-31 : sum of M=8..15 -> output row 2*tr + 1
    float s = ((acc[0] + acc[1]) + (acc[2] + acc[3]))
            + ((acc[4] + acc[5]) + (acc[6] + acc[7]));
    s *= 0.125f;  // (0.5)^3 Haar normalization

    const int orow = tr * 2 + h;
    out[(size_t)p * (OUT_W * OUT_W) + (size_t)orow * OUT_W + (tc * 16 + m)] = s;
}

extern "C" void kernel_launch(void* const* d_in, const int* in_sizes, int n_in,
                              void* d_out, int out_size, void* d_ws, size_t ws_size,
                              hipStream_t stream) {
    (void)in_sizes; (void)n_in; (void)out_size; (void)d_ws; (void)ws_size;
    const float* x = (const float*)d_in[0];
    float* out = (float*)d_out;

    // 96 planes * 512 tiles = 49152 waves; 8 waves (256 threads) per block
    dim3 block(32, 8, 1);
    dim3 grid((PLANES * TILES_PP) / 8, 1, 1);
    hipLaunchKernelGGL(haar3_wmma_kernel, grid, block, 0, stream, x, out);
}